// GravnetModel_73418170958212
// MI455X (gfx1250) — compile-verified
//
#include <hip/hip_runtime.h>
#include <math.h>

#define NTOT   20000
#define NPG    2500
#define DSH    32

typedef float v2f __attribute__((ext_vector_type(2)));
typedef float v8f __attribute__((ext_vector_type(8)));

__device__ __forceinline__ float eluf(float x) {
    return x > 0.f ? x : (expf(x) - 1.f);
}

// scalar 3-segment element load (only used by the post1 GEMM, K=67)
__device__ __forceinline__ float loadA3(const float* a0, int s0, int w0,
                                        const float* a1, int s1, int w1,
                                        const float* a2, int s2, int w2,
                                        int row, int k, int K) {
    const float* p = a0; int st = s0; int kk = k;
    if (k >= w0)      { p = a1; st = s1; kk = k - w0; }
    if (k >= w0 + w1) { p = a2; st = s2; kk = k - w0 - w1; }
    float v = 0.f;
    if (k < K) v = p[row * st + kk];
    return v;
}

// ---------------------------------------------------------------------------
// f32 WMMA GEMM, templated on:
//   NT   : number of 16-wide column tiles (compile-time -> no guards, NT wmmas)
//   NSEG : A segments (1 = plain, 2 = pointer-select + b64, 3 = scalar path)
//   KM4  : K % 4 == 0 (drop all K guards; A pair load is one aligned b64)
// One wave owns a 16-row stripe and all NT column tiles; the A fragment is
// software-pipelined (next K-step loaded while current wmmas issue).
// W staged in LDS transposed [col][Kp] so B is a single aligned ds_load_b64.
// ---------------------------------------------------------------------------
template <int NT, int NSEG, bool KM4>
__global__ __launch_bounds__(128) void gemm_wmma_t(
    const float* a0, int s0, int w0,
    const float* a1, int s1, int w1,
    const float* a2, int s2, int w2,
    const float* Wm, const float* bias,
    float* out, int outStride, int outColOff,
    int Nrows, int K, int M, int act)
{
    constexpr int Mp = NT * 16;
    __shared__ float WlT[Mp * 192];            // [col][Kp]

    const int tid  = threadIdx.x;
    const int wave = tid >> 5, lane = tid & 31;
    const int half = lane >> 4, mr = lane & 15;
    const int Kp   = KM4 ? K : ((K + 3) & ~3);

    // cooperative staged + transposed + zero-padded weight load
    for (int idx = tid; idx < Kp * Mp; idx += 128) {
        int kI = idx / Mp, m = idx - kI * Mp;
        float v = (kI < K && m < M) ? Wm[kI * M + m] : 0.f;
        WlT[m * Kp + kI] = v;
    }
    __syncthreads();

    const int nMT = (Nrows + 15) >> 4;
    const int mt  = blockIdx.x * 4 + wave;
    if (mt >= nMT) return;                     // wave-uniform, after the barrier
    const int row0 = mt << 4;
    const int arow = row0 + mr;

    const v2f* bp[NT];
    #pragma unroll
    for (int t = 0; t < NT; ++t) bp[t] = (const v2f*)(WlT + (t * 16 + mr) * Kp);

    v8f acc[NT];
    #pragma unroll
    for (int t = 0; t < NT; ++t) {
        v8f z = {0.f, 0.f, 0.f, 0.f, 0.f, 0.f, 0.f, 0.f};
        acc[t] = z;
    }

    auto loadPair = [&](int k0) -> v2f {
        if constexpr (NSEG == 1) {
            if constexpr (KM4) {
                return *(const v2f*)(a0 + arow * s0 + k0);   // aligned: stride & k0 even
            } else {
                v2f a;
                a.x = (k0     < K) ? a0[arow * s0 + k0]     : 0.f;
                a.y = (k0 + 1 < K) ? a0[arow * s0 + k0 + 1] : 0.f;
                return a;
            }
        } else if constexpr (NSEG == 2) {
            // w0 % 4 == 0 guaranteed by caller -> pair never straddles segments
            const float* p = a0; int st = s0, kk = k0;
            if (k0 >= w0) { p = a1; st = s1; kk = k0 - w0; }
            return *(const v2f*)(p + arow * st + kk);
        } else {
            v2f a;
            a.x = loadA3(a0, s0, w0, a1, s1, w1, a2, s2, w2, arow, k0,     K);
            a.y = loadA3(a0, s0, w0, a1, s1, w1, a2, s2, w2, arow, k0 + 1, K);
            return a;
        }
    };

    v2f aCur = loadPair(half * 2);
    for (int kk = 0; kk < K; kk += 4) {
        v2f aNext = {0.f, 0.f};
        if (kk + 4 < K) aNext = loadPair(kk + 4 + half * 2);   // pipeline next step
        const int bidx = (kk >> 1) + half;                     // v2f units in Kp row
        #pragma unroll
        for (int t = 0; t < NT; ++t)
            acc[t] = __builtin_amdgcn_wmma_f32_16x16x4_f32(
                false, aCur, false, bp[t][bidx], (short)0, acc[t], false, false);
        aCur = aNext;
    }

    // store: D layout VGPR r -> row (r + 8*half), col = lane%16 within tile
    #pragma unroll
    for (int t = 0; t < NT; ++t) {
        const int col = t * 16 + mr;
        if (col < M) {
            const float bv = bias ? bias[col] : 0.f;
            #pragma unroll
            for (int r = 0; r < 8; ++r) {
                const int rr = row0 + r + half * 8;
                if (rr < Nrows) {
                    float v = acc[t][r] + bv;
                    if (act == 1) v = eluf(v);
                    out[rr * outStride + outColOff + col] = v;
                }
            }
        }
    }
}

template <int NT, int NSEG, bool KM4>
static inline void rungemm(hipStream_t stream,
                           const float* A0, int S0, int W0,
                           const float* A1, int S1, int W1,
                           const float* A2, int S2, int W2,
                           const float* Wt, const float* Bs,
                           float* Out, int OS, int OC, int K, int M, int act)
{
    int nMT = (NTOT + 15) / 16;
    int blocks = (nMT + 3) / 4;
    gemm_wmma_t<NT, NSEG, KM4><<<blocks, 128, 0, stream>>>(
        A0, S0, W0, A1, S1, W1, A2, S2, W2, Wt, Bs, Out, OS, OC, NTOT, K, M, act);
}

// ---------------------------------------------------------------------------
// BatchNorm (training-mode batch stats): stats[c]=mean, stats[C+c]=rsqrt(var+eps)
// ---------------------------------------------------------------------------
__global__ __launch_bounds__(256) void bn_stats_k(const float* in, int stride,
                                                  int C, int Nrows, float* stats)
{
    __shared__ float ssum[256];
    __shared__ float ssq[256];
    const int c = blockIdx.x;
    float s = 0.f, q = 0.f;
    for (int r = threadIdx.x; r < Nrows; r += blockDim.x) {
        float v = in[r * stride + c];
        s += v; q += v * v;
    }
    ssum[threadIdx.x] = s; ssq[threadIdx.x] = q;
    __syncthreads();
    for (int off = 128; off > 0; off >>= 1) {
        if (threadIdx.x < off) {
            ssum[threadIdx.x] += ssum[threadIdx.x + off];
            ssq[threadIdx.x]  += ssq[threadIdx.x + off];
        }
        __syncthreads();
    }
    if (threadIdx.x == 0) {
        float m = ssum[0] / (float)Nrows;
        float var = ssq[0] / (float)Nrows - m * m;
        stats[c] = m;
        stats[C + c] = rsqrtf(var + 1e-5f);
    }
}

__global__ __launch_bounds__(256) void bn_apply_k(const float* in, int inStride,
                                                  float* out, int outStride, int outColOff,
                                                  const float* g, const float* b,
                                                  const float* stats, int C, int Nrows)
{
    const int idx = blockIdx.x * blockDim.x + threadIdx.x;
    if (idx >= Nrows * C) return;
    const int r = idx / C, c = idx % C;
    float v = in[r * inStride + c];
    out[r * outStride + outColOff + c] = (v - stats[c]) * stats[C + c] * g[c] + b[c];
}

// ---------------------------------------------------------------------------
// GravNet kNN + message aggregation. One wave per row; per-wave LDS slab holds
// all 2500 d^2 (float4 scans -> ds_load_b128). Exact k-smallest by iterated
// argmin with shuffle-xor reduce, index tie-break = lowest index. Then
// weighted mean/max over the k neighbors (lane = feature dim), with the
// data-dependent h-gather prefetched 8 neighbors ahead (global_prefetch_b8).
// ---------------------------------------------------------------------------
__global__ __launch_bounds__(128) void knn_agg_k(const float* sco, const float* h,
                                                 float* feat, int k, int Nrows)
{
    __shared__ float d2s[4][NPG];              // rows 16B aligned (10000B)
    __shared__ int   sidx[4][256];
    __shared__ float swei[4][256];
    const int w = threadIdx.x >> 5;
    const int lane = threadIdx.x & 31;
    const int row = blockIdx.x * 4 + w;
    const bool active = row < Nrows;           // wave-uniform
    const int base = (active ? row / NPG : 0) * NPG;

    if (active) {
        const float sx = sco[row * 3 + 0];
        const float sy = sco[row * 3 + 1];
        const float sz = sco[row * 3 + 2];
        for (int c = lane; c < NPG; c += 32) {
            float dx = sco[(base + c) * 3 + 0] - sx;
            float dy = sco[(base + c) * 3 + 1] - sy;
            float dz = sco[(base + c) * 3 + 2] - sz;
            d2s[w][c] = dx * dx + dy * dy + dz * dz;
        }
    }
    __syncthreads();

    const float4* dv = (const float4*)&d2s[w][0];
    for (int sel = 0; sel < k; ++sel) {
        float best = 3.0e38f;
        int bi = NPG;
        if (active) {
            for (int q = lane; q < NPG / 4; q += 32) {   // 625 float4 chunks
                float4 v = dv[q];
                int c = q << 2;
                if (v.x < best || (v.x == best && c     < bi)) { best = v.x; bi = c;     }
                if (v.y < best || (v.y == best && c + 1 < bi)) { best = v.y; bi = c + 1; }
                if (v.z < best || (v.z == best && c + 2 < bi)) { best = v.z; bi = c + 2; }
                if (v.w < best || (v.w == best && c + 3 < bi)) { best = v.w; bi = c + 3; }
            }
        }
        #pragma unroll
        for (int off = 16; off > 0; off >>= 1) {
            float ov = __shfl_xor(best, off, 32);
            int   oi = __shfl_xor(bi, off, 32);
            if (ov < best || (ov == best && oi < bi)) { best = ov; bi = oi; }
        }
        if (active && lane == 0) {
            sidx[w][sel] = bi;
            swei[w][sel] = expf(-10.f * best);           // w = exp(10 * (-d^2))
            d2s[w][bi] = 3.0e38f;
        }
        __syncthreads();
    }

    if (active) {
        float sum = 0.f, mx = -3.0e38f;
        for (int j = 0; j < k; ++j) {
            if (j + 8 < k) {
                int pid = sidx[w][j + 8];
                __builtin_prefetch(&h[(base + pid) * DSH + lane], 0, 1);
            }
            const int id = sidx[w][j];
            const float wt = swei[w][j];
            float v = h[(base + id) * DSH + lane] * wt;
            sum += v;
            mx = fmaxf(mx, v);
        }
        feat[row * (2 * DSH) + lane]       = sum / (float)k;
        feat[row * (2 * DSH) + DSH + lane] = mx;
    }
}

// ---------------------------------------------------------------------------
// Output heads: cc(3), beta(1), energy(1). d_out = [N*4 | N*1] flat f32.
// ---------------------------------------------------------------------------
__global__ __launch_bounds__(256) void heads_k(const float* z,
                                               const float* Wc,
                                               const float* Wb, const float* bb,
                                               const float* We,
                                               float* out, int Nrows)
{
    const int row = blockIdx.x * blockDim.x + threadIdx.x;
    if (row >= Nrows) return;
    const float* zr = z + row * 64;
    float c0 = 0.f, c1 = 0.f, c2 = 0.f, be = bb[0], en = 0.f;
    for (int i = 0; i < 64; ++i) {
        float zi = zr[i];
        c0 += zi * Wc[i * 3 + 0];
        c1 += zi * Wc[i * 3 + 1];
        c2 += zi * Wc[i * 3 + 2];
        be += zi * Wb[i];
        en += zi * We[i];
    }
    out[row * 4 + 0] = c0;
    out[row * 4 + 1] = c1;
    out[row * 4 + 2] = c2;
    out[row * 4 + 3] = be;
    out[Nrows * 4 + row] = 1.f + eluf(en);
}

// ---------------------------------------------------------------------------
extern "C" void kernel_launch(void* const* d_in, const int* in_sizes, int n_in,
                              void* d_out, int out_size, void* d_ws, size_t ws_size,
                              hipStream_t stream)
{
    (void)in_sizes; (void)n_in; (void)out_size; (void)ws_size;
    const int Nn = NTOT;
    const int KS_[4] = {16, 128, 16, 256};
    const int BIN[4] = {64, 96, 128, 160};

    auto IN = [&](int i) -> const float* { return (const float*)d_in[i]; };
    // input layout (setup_inputs insertion order):
    // 0: x, 1: bn1.g, 2: bn1.b, 3: dense1_W
    // block i base = 4 + 18*i:
    //   +0/1 pre1.W/b  +2/3 pre2.W/b  +4/5 bn_a.g/b  +6/7 lin_s.W/b
    //   +8/9 lin_h.W/b +10/11 lin_out.W/b +12/13 post1.W/b +14/15 post2.W/b
    //   +16/17 bn_b.g/b
    // 76..81 postgn {W,b}x3, 82/83 bn2.g/b, 84 clust.W, 85/86 beta.W/b, 87 energy.W
    const int BB = 4;

    // scratch layout
    float* W = (float*)d_ws;
    size_t off = 0;
    auto alloc = [&](size_t n) { float* p = W + off; off += n; return p; };
    float* xbn   = alloc((size_t)Nn * 9);
    float* feats = alloc((size_t)Nn * 192);
    float* t1    = alloc((size_t)Nn * 32);
    float* t2    = alloc((size_t)Nn * 32);
    float* xa    = alloc((size_t)Nn * 32);
    float* sb    = alloc((size_t)Nn * 3);
    float* hb    = alloc((size_t)Nn * 32);
    float* fk    = alloc((size_t)Nn * 64);
    float* xg    = alloc((size_t)Nn * 32);
    float* u1    = alloc((size_t)Nn * 32);
    float* u2    = alloc((size_t)Nn * 32);
    float* z1    = alloc((size_t)Nn * 64);
    float* z2    = alloc((size_t)Nn * 64);
    float* z3    = alloc((size_t)Nn * 64);
    float* stats = alloc(512);

    auto bnorm = [&](const float* in, int is, float* out, int os, int oc,
                     const float* g, const float* b, int C) {
        bn_stats_k<<<C, 256, 0, stream>>>(in, is, C, Nn, stats);
        int th = Nn * C;
        bn_apply_k<<<(th + 255) / 256, 256, 0, stream>>>(in, is, out, os, oc,
                                                         g, b, stats, C, Nn);
    };
    const float* nl = nullptr;

    // h = bn1(x) @ dense1_W  -> feats[:, 0:64]   (K=9 not mul4, M=64)
    bnorm(IN(0), 9, xbn, 9, 0, IN(1), IN(2), 9);
    rungemm<4, 1, false>(stream, xbn, 9, 9, nl, 0, 0, nl, 0, 0, IN(3), nl,
                         feats, 192, 0, 9, 64, 0);

    for (int i = 0; i < 4; ++i) {
        const int p = BB + 18 * i;
        const int fin = BIN[i];
        const int k = KS_[i];
        // pre1 -> elu -> pre2 -> elu -> bn_a
        rungemm<2, 1, true>(stream, feats, 192, fin, nl, 0, 0, nl, 0, 0,
                            IN(p + 0), IN(p + 1), t1, 32, 0, fin, 32, 1);
        rungemm<2, 1, true>(stream, t1, 32, 32, nl, 0, 0, nl, 0, 0,
                            IN(p + 2), IN(p + 3), t2, 32, 0, 32, 32, 1);
        bnorm(t2, 32, xa, 32, 0, IN(p + 4), IN(p + 5), 32);
        // learned coords + features to propagate
        rungemm<1, 1, true>(stream, xa, 32, 32, nl, 0, 0, nl, 0, 0,
                            IN(p + 6), IN(p + 7), sb, 3, 0, 32, 3, 0);
        rungemm<2, 1, true>(stream, xa, 32, 32, nl, 0, 0, nl, 0, 0,
                            IN(p + 8), IN(p + 9), hb, 32, 0, 32, 32, 0);
        // kNN + weighted mean/max aggregation
        knn_agg_k<<<(Nn + 3) / 4, 128, 0, stream>>>(sb, hb, fk, k, Nn);
        // out = [xa | feat] @ lin_out   (2 segments, w0=32 (mult of 4), K=96)
        rungemm<2, 2, true>(stream, xa, 32, 32, fk, 64, 64, nl, 0, 0,
                            IN(p + 10), IN(p + 11), xg, 32, 0, 96, 32, 0);
        // post: [xgn | s | xa] (K=67, 3 segments) -> post1 elu -> post2 elu -> bn_b
        rungemm<2, 3, false>(stream, xg, 32, 32, sb, 3, 3, xa, 32, 32,
                             IN(p + 12), IN(p + 13), u1, 32, 0, 67, 32, 1);
        rungemm<2, 1, true>(stream, u1, 32, 32, nl, 0, 0, nl, 0, 0,
                            IN(p + 14), IN(p + 15), u2, 32, 0, 32, 32, 1);
        bnorm(u2, 32, feats, 192, 64 + 32 * i, IN(p + 16), IN(p + 17), 32);
    }

    // post-GN trunk: 192 -> 64 -> 64 -> 64 (all ELU), then bn2
    rungemm<4, 1, true>(stream, feats, 192, 192, nl, 0, 0, nl, 0, 0,
                        IN(76), IN(77), z1, 64, 0, 192, 64, 1);
    rungemm<4, 1, true>(stream, z1, 64, 64, nl, 0, 0, nl, 0, 0,
                        IN(78), IN(79), z2, 64, 0, 64, 64, 1);
    rungemm<4, 1, true>(stream, z2, 64, 64, nl, 0, 0, nl, 0, 0,
                        IN(80), IN(81), z3, 64, 0, 64, 64, 1);
    bnorm(z3, 64, z3, 64, 0, IN(82), IN(83), 64);

    // heads -> d_out ([N,4] cc|beta, then [N,1] energy)
    heads_k<<<(Nn + 255) / 256, 256, 0, stream>>>(z3, IN(84), IN(85), IN(86),
                                                  IN(87), (float*)d_out, Nn);
}